// ShiftedWindowMSA_25933012533481
// MI455X (gfx1250) — compile-verified
//
#include <hip/hip_runtime.h>
#include <math.h>

// ---------------------------------------------------------------------------
// Fully-fused shifted-window MSA for gfx1250 (MI455X), fp32 WMMA.
//
// One workgroup (128 threads = 4 wave32) per (batch, window), ~302.8 KB of the
// 320 KB CDNA5 WGP LDS:
//   - x-panel and attention-output panel both resident (zero intermediate HBM)
//   - DOUBLE-BUFFERED weight chunks staged with GLOBAL_LOAD_ASYNC_TO_LDS_B128:
//     software pipeline issues chunk k+1 while chunk k's 96 WMMAs execute;
//     the next head's first chunk prefetches across the whole attention phase
//   - all four GEMM stages on V_WMMA_F32_16X16X4_F32 (full fp32 fidelity)
//   - roll(-4/+4), rel-sin embedding (13x13 table), shift masks as index math
// ---------------------------------------------------------------------------

typedef __attribute__((ext_vector_type(2))) float v2f;
typedef __attribute__((ext_vector_type(8))) float v8f;

#define NTHREADS 128

// LDS strides in floats (bank-conflict-free for the WMMA operand patterns)
#define LDA  388   // 384 + 4 : panels (388 % 64 == 4 -> disjoint half-wave banks)
#define LDB  104   // 96  + 8 : weight chunks (2*104 % 64 == 16 -> disjoint)
#define LDK  34    // 32  + 2 : Q/K tiles [token][e]
#define LDV  68    // 64  + 4 : V^T tile [e][token]
#define LDSS 68    // 64  + 4 : score tile [q][k]

// LDS float offsets (all multiples of 4 -> 16B-aligned async destinations)
#define OFF_XP   0
#define OFF_AP   (64 * LDA)                 // attention output panel
#define OFF_B0   (2 * 64 * LDA)             // weight chunk buffer 0
#define OFF_B1   (OFF_B0 + 64 * LDB)        // weight chunk buffer 1
#define OFF_SS   (OFF_B1 + 64 * LDB)        // scores (own region now)
#define OFF_QS   (OFF_SS + 64 * LDSS)
#define OFF_KS   (OFF_QS + 64 * LDK)
#define OFF_VT   (OFF_KS + 64 * LDK)
#define OFF_RED  (OFF_VT + 32 * LDV)        // 128-float softmax scratch
#define OFF_BTAB (OFF_RED + 128)            // 13x13 embedding table (+pad)
#define OFF_BQ   (OFF_BTAB + 176)           // b_qkv copy (1152)
#define OFF_BO   (OFF_BQ + 1152)            // b_out copy (384)
#define LDS_FLOATS (OFF_BO + 384)           // = 75696 floats = 302,784 B

static __device__ __forceinline__ v8f wmma4(v2f a, v2f b, v8f c) {
  return __builtin_amdgcn_wmma_f32_16x16x4_f32(false, a, false, b, (short)0, c,
                                               false, false);
}
static __device__ __forceinline__ v8f zero8() {
  v8f z = {0.f, 0.f, 0.f, 0.f, 0.f, 0.f, 0.f, 0.f};
  return z;
}
// A operand (M=16,K=4): lanes 0-15 rows @K{k,k+1}, lanes 16-31 @K{k+2,k+3}
static __device__ __forceinline__ v2f load_a(const float* s, int ld, int mbase,
                                             int k, int lane) {
  const float* p = s + (mbase + (lane & 15)) * ld + k + ((lane >> 4) << 1);
  v2f r; r.x = p[0]; r.y = p[1]; return r;
}
// B operand (K=4,N=16) from row-major [k][n]
static __device__ __forceinline__ v2f load_b(const float* s, int ld, int nbase,
                                             int k, int lane) {
  const float* p = s + (k + ((lane >> 4) << 1)) * ld + nbase + (lane & 15);
  v2f r; r.x = p[0]; r.y = p[ld]; return r;
}
// B operand (K=4,N=16) from [n][k] storage (K^T from K tile, V^T tile)
static __device__ __forceinline__ v2f load_bt(const float* s, int ld, int nbase,
                                              int k, int lane) {
  const float* p = s + (nbase + (lane & 15)) * ld + k + ((lane >> 4) << 1);
  v2f r; r.x = p[0]; r.y = p[1]; return r;
}

// CDNA5 async global -> LDS copy (ASYNCcnt-tracked, no VGPR round trip).
// lds_byte_off = byte offset inside this WG's LDS allocation (all-dynamic).
static __device__ __forceinline__ void async_b128(unsigned lds_byte_off,
                                                  const float* g) {
  asm volatile("global_load_async_to_lds_b128 %0, %1, off"
               :: "v"(lds_byte_off), "v"(g)
               : "memory");
}
static __device__ __forceinline__ void wait_async() {
  asm volatile("s_wait_asynccnt 0x0" ::: "memory");
}

// Issue async staging for linear weight chunk c (0..71: qkv, 72..95: out-proj)
// into double buffer (c & 1). Each chunk is a 64(K) x 96(N) fp32 slab.
static __device__ __forceinline__ void issue_chunk(int c, int tid,
                                                   const float* __restrict__ wqkv,
                                                   const float* __restrict__ wout) {
  const float* src;
  size_t rstride;
  int colbase, kbase;
  if (c < 72) {
    src = wqkv; rstride = 1152; colbase = 96 * (c / 6); kbase = 64 * (c % 6);
  } else {
    int d = c - 72;
    src = wout; rstride = 384; colbase = 96 * (d / 6); kbase = 64 * (d % 6);
  }
  const unsigned bufoff = (c & 1) ? OFF_B1 : OFF_B0;
  for (int idx = tid; idx < 64 * 24; idx += NTHREADS) {
    int row = idx / 24, c4 = (idx % 24) * 4;
    async_b128((bufoff + (unsigned)(row * LDB + c4)) * 4u,
               src + (size_t)(kbase + row) * rstride + colbase + c4);
  }
}

// 96 WMMAs: acc[0..5] += Apanel[mb.., kbase..kbase+63] * B[64 x 96]
static __device__ __forceinline__ void gemm_chunk(const float* __restrict__ Ap,
                                                  const float* __restrict__ B,
                                                  int kbase, int mb, int lane,
                                                  v8f acc[6]) {
#pragma unroll
  for (int kk = 0; kk < 16; ++kk) {
    v2f a = load_a(Ap, LDA, mb, kbase + kk * 4, lane);
    v2f bb[6];
#pragma unroll
    for (int nt = 0; nt < 6; ++nt) bb[nt] = load_b(B, LDB, nt * 16, kk * 4, lane);
#pragma unroll
    for (int nt = 0; nt < 6; ++nt) acc[nt] = wmma4(a, bb[nt], acc[nt]);
  }
}

extern "C" __global__ void __launch_bounds__(NTHREADS)
swin_msa_fused(const float* __restrict__ x,
               const float* __restrict__ wqkv,
               const float* __restrict__ bqkv,
               const float* __restrict__ wout,
               const float* __restrict__ bout,
               float* __restrict__ out) {
  extern __shared__ float sm[];
  float* Xp   = sm + OFF_XP;
  float* Apan = sm + OFF_AP;
  float* Ss   = sm + OFF_SS;
  float* Qs   = sm + OFF_QS;
  float* Ks   = sm + OFF_KS;
  float* Vt   = sm + OFF_VT;
  float* Red  = sm + OFF_RED;
  float* Btab = sm + OFF_BTAB;
  float* BQl  = sm + OFF_BQ;
  float* BOl  = sm + OFF_BO;

  const int blk = blockIdx.x;  // b*64 + win
  const int b   = blk >> 6;
  const int win = blk & 63;
  const int wy  = win >> 3, wx = win & 7;
  const int tid = threadIdx.x;
  const int lane = tid & 31, wv = tid >> 5;
  const int mb = 16 * wv;

  // ---- init: zero pad rows (49..63) of both panels (plain DS stores)
  for (int idx = tid; idx < 15 * 384; idx += NTHREADS) {
    int r = 49 + idx / 384, c = idx % 384;
    Xp[r * LDA + c] = 0.f;
    Apan[r * LDA + c] = 0.f;
  }
  // ---- async-stage x panel rows < 49 (roll(-4) folded into the gather)
  for (int idx = tid; idx < 49 * 96; idx += NTHREADS) {
    int row = idx / 96, c4 = (idx % 96) * 4;
    int m1 = row / 7, m2 = row % 7;
    int i = (wy * 7 + m1 + 4) % 56;
    int j = (wx * 7 + m2 + 4) % 56;
    async_b128((unsigned)((OFF_XP + row * LDA + c4) * 4),
               x + (((size_t)b * 56 + i) * 56 + j) * 384 + c4);
  }
  // ---- async-stage bias vectors
  for (int idx = tid; idx < 72 + 24; idx += NTHREADS) {
    if (idx < 72) async_b128((OFF_BQ + (unsigned)idx * 4) * 4u, bqkv + idx * 4);
    else {
      int d = idx - 72;
      async_b128((OFF_BO + (unsigned)d * 4) * 4u, bout + d * 4);
    }
  }
  // ---- relative sinusoidal embedding base table B[13][13]
  if (tid < 169) {
    int r = tid / 13, c = tid % 13;
    float divisor = expf((float)(2 * (c >> 1)) * (-logf(10000.0f) / 13.0f));
    float ang = (float)r * divisor;
    Btab[tid] = (c & 1) ? cosf(ang) : sinf(ang);
  }
  // ---- prime the weight-chunk pipeline
  issue_chunk(0, tid, wqkv, wout);

  const float qscale = 0.17677669529663687f; // 1/sqrt(32)
  int chunk = 0;                             // 0..71 qkv, 72..95 out-proj

  // ========================== per-head attention ==========================
  for (int hh = 0; hh < 12; ++hh) {
    v8f acc[6];
#pragma unroll
    for (int t = 0; t < 6; ++t) acc[t] = zero8();
    const int nb = 96 * hh;

    for (int kc = 0; kc < 6; ++kc) {
      wait_async();                // chunk `chunk` landed (wave's own issues)
      __syncthreads();             // ...and everyone else's; prev buffer free
      if (chunk + 1 < 96) issue_chunk(chunk + 1, tid, wqkv, wout);
      gemm_chunk(Xp, sm + ((chunk & 1) ? OFF_B1 : OFF_B0), 64 * kc, mb, lane, acc);
      ++chunk;
    }
    // (next head's / out-proj's first chunk is now in flight: it prefetches
    //  across the entire attention phase below)

    // ----- epilogue: bias, Q-scale, de-interleave into Q/K/V^T LDS tiles
#pragma unroll
    for (int nt = 0; nt < 6; ++nt) {
#pragma unroll
      for (int v = 0; v < 8; ++v) {
        int m = mb + v + ((lane >> 4) << 3);
        int n = nt * 16 + (lane & 15);     // 0..95
        float val = acc[nt][v] + BQl[nb + n];
        int e = n / 3, s = n - 3 * e;
        if (s == 0)      Qs[m * LDK + e] = val * qscale;
        else if (s == 1) Ks[m * LDK + e] = val;
        else             Vt[e * LDV + m] = val;
      }
    }
    __syncthreads();

    // ----- S = Q * K^T  (64x64, K=32)
    v8f sacc[4];
#pragma unroll
    for (int t = 0; t < 4; ++t) sacc[t] = zero8();
#pragma unroll
    for (int kk = 0; kk < 8; ++kk) {
      v2f a = load_a(Qs, LDK, mb, kk * 4, lane);
      v2f bb[4];
#pragma unroll
      for (int nt = 0; nt < 4; ++nt) bb[nt] = load_bt(Ks, LDK, nt * 16, kk * 4, lane);
#pragma unroll
      for (int nt = 0; nt < 4; ++nt) sacc[nt] = wmma4(a, bb[nt], sacc[nt]);
    }
    // ----- embedding + shift masks + pad masks -> Ss
#pragma unroll
    for (int nt = 0; nt < 4; ++nt) {
#pragma unroll
      for (int v = 0; v < 8; ++v) {
        int q = mb + v + ((lane >> 4) << 3);
        int k = nt * 16 + (lane & 15);
        float sv;
        if (q < 49 && k < 49) {
          sv = sacc[nt][v];
          int xi = (k - q) / 7;            // trunc-toward-zero == .int()
          int yi = (k % 7) - (q % 7);
          int bi = (xi < 0) ? xi + 13 : xi;
          int bj = (yi < 0) ? yi + 13 : yi;
          sv += Btab[bi * 13 + bj];
          if (wy == 7 && (((q / 7) >= 4) ^ ((k / 7) >= 4))) sv = -1e30f;
          if (wx == 7 && (((q % 7) >= 4) ^ ((k % 7) >= 4))) sv = -1e30f;
        } else {
          sv = (q < 49) ? -1e30f : 0.0f;   // pad keys masked; pad q rows inert
        }
        Ss[q * LDSS + k] = sv;
      }
    }
    __syncthreads();
    // ----- softmax over k: 2 threads per row, 32 cols each
    {
      int r = tid & 63, hf = tid >> 6;
      float* row = Ss + r * LDSS + 32 * hf;
      float mx = row[0];
      for (int k = 1; k < 32; ++k) mx = fmaxf(mx, row[k]);
      Red[tid] = mx;
      __syncthreads();
      float m2 = fmaxf(Red[r], Red[r + 64]);
      __syncthreads();
      float s = 0.f;
      for (int k = 0; k < 32; ++k) {
        float e = __expf(row[k] - m2);
        row[k] = e;
        s += e;
      }
      Red[tid] = s;
      __syncthreads();
      float inv = 1.0f / (Red[r] + Red[r + 64]);
      for (int k = 0; k < 32; ++k) row[k] *= inv;
    }
    __syncthreads();

    // ----- O = P * V (64x32, K=64); result goes straight into att panel
    v8f oacc[2];
    oacc[0] = zero8(); oacc[1] = zero8();
#pragma unroll
    for (int kk = 0; kk < 16; ++kk) {
      v2f a = load_a(Ss, LDSS, mb, kk * 4, lane);
      v2f bb[2];
#pragma unroll
      for (int nt = 0; nt < 2; ++nt) bb[nt] = load_bt(Vt, LDV, nt * 16, kk * 4, lane);
#pragma unroll
      for (int nt = 0; nt < 2; ++nt) oacc[nt] = wmma4(a, bb[nt], oacc[nt]);
    }
#pragma unroll
    for (int nt = 0; nt < 2; ++nt) {
#pragma unroll
      for (int v = 0; v < 8; ++v) {
        int q = mb + v + ((lane >> 4) << 3);
        if (q < 49) {
          int e = nt * 16 + (lane & 15);
          Apan[q * LDA + hh * 32 + e] = oacc[nt][v];
        }
      }
    }
  }

  // ========================== output projection ==========================
  for (int nc = 0; nc < 4; ++nc) {           // 384 cols in 4 x 96 chunks
    v8f acc[6];
#pragma unroll
    for (int t = 0; t < 6; ++t) acc[t] = zero8();
    const int nb = 96 * nc;

    for (int kc = 0; kc < 6; ++kc) {
      wait_async();
      __syncthreads();
      if (chunk + 1 < 96) issue_chunk(chunk + 1, tid, wqkv, wout);
      gemm_chunk(Apan, sm + ((chunk & 1) ? OFF_B1 : OFF_B0), 64 * kc, mb, lane, acc);
      ++chunk;
    }
    // epilogue: bias + un-window + roll(+4) scatter to d_out
#pragma unroll
    for (int nt = 0; nt < 6; ++nt) {
#pragma unroll
      for (int v = 0; v < 8; ++v) {
        int m = mb + v + ((lane >> 4) << 3);
        if (m < 49) {
          int n = nb + nt * 16 + (lane & 15);
          int m1 = m / 7, m2 = m % 7;
          int i = (wy * 7 + m1 + 4) % 56;
          int j = (wx * 7 + m2 + 4) % 56;
          out[(((size_t)b * 56 + i) * 56 + j) * 384 + n] = acc[nt][v] + BOl[n];
        }
      }
    }
  }
}

extern "C" void kernel_launch(void* const* d_in, const int* in_sizes, int n_in,
                              void* d_out, int out_size, void* d_ws,
                              size_t ws_size, hipStream_t stream) {
  (void)in_sizes; (void)n_in; (void)out_size; (void)d_ws; (void)ws_size;
  const float* x    = (const float*)d_in[0];
  const float* wqkv = (const float*)d_in[1];
  const float* bqkv = (const float*)d_in[2];
  const float* wout = (const float*)d_in[3];
  const float* bout = (const float*)d_in[4];
  float* out = (float*)d_out;

  const size_t lds = LDS_FLOATS * sizeof(float); // ~302.8 KB of 320 KB WGP LDS
  const int nblk = 16 * 64;                      // one WG per (batch, window)
  swin_msa_fused<<<nblk, NTHREADS, lds, stream>>>(x, wqkv, bqkv, wout, bout, out);
}